// S4DKernel_63170378990065
// MI455X (gfx1250) — compile-verified
//
#include <hip/hip_runtime.h>
#include <math.h>

// S4D Vandermonde kernel K[d,l] = Re( sum_n CB[d,n] * Abar[d,n]^l )
// Factorized per 256-chunk of l into a 16x64 x 64x16 f32 GEMM done with
// 16 chained V_WMMA_F32_16X16X4_F32 ops (wave32, full EXEC).

typedef __attribute__((ext_vector_type(2))) float v2f;
typedef __attribute__((ext_vector_type(8))) float v8f;

#define D_MODEL 1024
#define D_STATE 32
#define SEQ_LEN 4096

__device__ __forceinline__ float softplus_f(float x) {
    // inputs here are <= ~0, log1p(exp(x)) is stable
    return log1pf(__expf(x));
}

__global__ __launch_bounds__(32) void s4d_wmma_kernel(
    const float* __restrict__ log_A_real,  // (D_MODEL, D_STATE)
    const float* __restrict__ A_imag,      // (D_MODEL, D_STATE)
    const float* __restrict__ B,           // (D_MODEL, D_STATE, 2)
    const float* __restrict__ C,           // (D_MODEL, D_STATE, 2)
    const float* __restrict__ log_dt,      // (D_MODEL,)
    float* __restrict__ K)                 // (D_MODEL, SEQ_LEN)
{
    const int lane = threadIdx.x;          // one full wave32; EXEC all-1s for WMMA
    const int hi   = lane >> 4;            // 0: lanes 0-15, 1: lanes 16-31
    const int m    = lane & 15;            // A row i, B col j, D col N
    const int d    = blockIdx.y;
    const int l0   = blockIdx.x << 8;      // 256 outputs per workgroup

    const float dt = softplus_f(log_dt[d]);

    // Per-lane slices of U (16x32 cplx) and V (32x16 cplx, imag pre-negated).
    // Lane covers states n = 4t + 2*hi + p for t=0..7, p=0..1, matching the
    // f32 WMMA A/B VGPR layout: K-index within a 4-chunk = p + 2*hi.
    float Ure[8][2], Uim[8][2], Vre[8][2], Vim[8][2];

    const float lU = (float)(l0 + 16 * m); // power for U row i = m
    const float lV = (float)m;             // power for V col j = m

    #pragma unroll
    for (int t = 0; t < 8; ++t) {
        #pragma unroll
        for (int p = 0; p < 2; ++p) {
            const int n  = 4 * t + 2 * hi + p;
            const int dn = d * D_STATE + n;
            const float ar  = -softplus_f(log_A_real[dn]); // forced negative
            const float ai  = A_imag[dn];
            const float dar = dt * ar;                     // Re(log Abar)
            const float dai = dt * ai;                     // Im(log Abar)
            const float br = B[2 * dn + 0], bi = B[2 * dn + 1];
            const float cr = C[2 * dn + 0], ci = C[2 * dn + 1];
            const float cbr = cr * br - ci * bi;           // CB = C*B (complex)
            const float cbi = cr * bi + ci * br;

            // U[i,n] = CB_n * exp(dtA_n * (l0 + 16*i))
            float su, cu;
            __sincosf(dai * lU, &su, &cu);
            const float eu = __expf(dar * lU);
            const float pr = eu * cu, pi = eu * su;
            Ure[t][p] = cbr * pr - cbi * pi;
            Uim[t][p] = cbr * pi + cbi * pr;

            // V[n,j] = exp(dtA_n * j); store imag negated so the second
            // WMMA pass accumulates -Ui*Vi (f32 WMMA has no A/B neg bits)
            float sv, cv;
            __sincosf(dai * lV, &sv, &cv);
            const float ev = __expf(dar * lV);
            Vre[t][p] = ev * cv;
            Vim[t][p] = -(ev * sv);
        }
    }

    // D = Ur*Vr + Uim*(-Vi)  : 16 chained 16x16x4 f32 WMMAs (K = 64 total)
    v8f acc = {0.f, 0.f, 0.f, 0.f, 0.f, 0.f, 0.f, 0.f};
    #pragma unroll
    for (int t = 0; t < 8; ++t) {
        v2f a, b;
        a.x = Ure[t][0]; a.y = Ure[t][1];
        b.x = Vre[t][0]; b.y = Vre[t][1];
        acc = __builtin_amdgcn_wmma_f32_16x16x4_f32(false, a, false, b,
                                                    (short)0, acc, false, false);
    }
    #pragma unroll
    for (int t = 0; t < 8; ++t) {
        v2f a, b;
        a.x = Uim[t][0]; a.y = Uim[t][1];
        b.x = Vim[t][0]; b.y = Vim[t][1];
        acc = __builtin_amdgcn_wmma_f32_16x16x4_f32(false, a, false, b,
                                                    (short)0, acc, false, false);
    }

    // C/D layout: VGPR r -> M = r + 8*hi, N = m ; l = l0 + 16*M + N
    float* out = K + (size_t)d * SEQ_LEN + l0;
    #pragma unroll
    for (int r = 0; r < 8; ++r) {
        const int M = r + 8 * hi;
        out[16 * M + m] = acc[r];
    }
}

extern "C" void kernel_launch(void* const* d_in, const int* in_sizes, int n_in,
                              void* d_out, int out_size, void* d_ws, size_t ws_size,
                              hipStream_t stream) {
    const float* log_A_real = (const float*)d_in[0];
    const float* A_imag     = (const float*)d_in[1];
    const float* B          = (const float*)d_in[2];
    const float* C          = (const float*)d_in[3];
    const float* log_dt     = (const float*)d_in[4];
    (void)in_sizes; (void)n_in; (void)d_ws; (void)ws_size; (void)out_size;

    dim3 grid(SEQ_LEN / 256, D_MODEL);  // (16, 1024): one wave per (l-block, d)
    dim3 block(32);                     // exactly one wave32
    s4d_wmma_kernel<<<grid, block, 0, stream>>>(log_A_real, A_imag, B, C, log_dt,
                                                (float*)d_out);
}